// LocalContrastEnhancement_10453950398757
// MI455X (gfx1250) — compile-verified
//
#include <hip/hip_runtime.h>
#include <math.h>

typedef float v2f __attribute__((ext_vector_type(2)));
typedef float v8f __attribute__((ext_vector_type(8)));

// Geometry
#define IMG_W       1024
#define TILE_W      64
#define TILE_H      32
#define HALO        7
#define IN_ROWS     (TILE_H + 2 * HALO)   // 46 input rows staged
#define TILE_STRIDE 84                    // 80 cols used, padded (mult of 4 dwords for B128, 2-way banks max)
#define H_STRIDE    65                    // 64 cols + 1 pad -> conflict-free column access
#define TILE_DW     (IN_ROWS * TILE_STRIDE)          // 3864
#define HX_OFF      TILE_DW                          // 3864
#define HXX_OFF     (HX_OFF + IN_ROWS * H_STRIDE)    // 6854
#define SMEM_DW     (HXX_OFF + IN_ROWS * H_STRIDE)   // 9844 dwords = 39376 B
#define NQUADS      20                    // 20 x B128 per staged row (80 floats)

__global__ __launch_bounds__(256)
void lce_box15_kernel(const float* __restrict__ x, float* __restrict__ out) {
    extern __shared__ float smem[];
    const int tid   = threadIdx.x;
    const int col0  = blockIdx.x * TILE_W;
    const int row0  = blockIdx.y * TILE_H;
    const float* xp = x   + (size_t)blockIdx.z * (IMG_W * IMG_W);
    float*       op = out + (size_t)blockIdx.z * (IMG_W * IMG_W);

    // ---- Phase 0: zero staging tile (gives us exact zero padding at image borders)
    for (int i = tid; i < TILE_DW; i += 256) smem[i] = 0.0f;
    __syncthreads();   // dscnt waited by barrier before async engine writes LDS

    // ---- Phase A: async global -> LDS, 16B-aligned B128 quads.
    // Tile col 0 maps to global col (col0 - 8) so quads are aligned; OOB is
    // all-or-nothing per quad (1024 and col0-8 are multiples of 4).
    for (int j = tid; j < IN_ROWS * NQUADS; j += 256) {
        const int r    = j / NQUADS;
        const int q    = j % NQUADS;
        const int grow = row0 - HALO + r;
        const int gcol = col0 - 8 + q * 4;
        if (grow >= 0 && grow < IMG_W && gcol >= 0 && gcol + 3 < IMG_W) {
            const unsigned lds_off = (unsigned)((r * TILE_STRIDE + q * 4) * 4);
            const float* gp = xp + (size_t)grow * IMG_W + gcol;
            asm volatile("global_load_async_to_lds_b128 %0, %1, off"
                         :: "v"(lds_off), "v"(gp) : "memory");
        }
    }
    asm volatile("s_wait_asynccnt 0x0" ::: "memory");
    __syncthreads();

    // ---- Phase B: horizontal 15-tap sums of x and x^2, sliding window in registers.
    // Job = (input row r, 16-col segment s). Output col c=16s+j uses tile cols c+1..c+15.
    for (int j = tid; j < IN_ROWS * 4; j += 256) {
        const int r    = j >> 2;
        const int s    = j & 3;
        const int base = r * TILE_STRIDE + s * 16 + 1;
        float v[30];
        #pragma unroll
        for (int t = 0; t < 30; ++t) v[t] = smem[base + t];
        float sx = 0.0f, sxx = 0.0f;
        #pragma unroll
        for (int t = 0; t < 15; ++t) { sx += v[t]; sxx += v[t] * v[t]; }
        const int hb = r * H_STRIDE + s * 16;
        smem[HX_OFF  + hb] = sx;
        smem[HXX_OFF + hb] = sxx;
        #pragma unroll
        for (int t = 1; t < 16; ++t) {
            const float ad = v[t + 14], sb = v[t - 1];
            sx  += ad - sb;
            sxx += ad * ad - sb * sb;
            smem[HX_OFF  + hb + t] = sx;
            smem[HXX_OFF + hb + t] = sxx;
        }
    }
    __syncthreads();

    // ---- Phase C: vertical 15-tap sum as WMMA: D(16x16) = Band(16x4-chunks) x Hsum(4x16).
    // A layout (documented): lanes 0-15 M=lane, VGPR0/1 = K0/K1; lanes 16-31 = K2/K3.
    // B layout (mirrored): lanes 0-15 N=lane hold K0/K1; lanes 16-31 hold K2/K3.
    // D layout (documented): VGPR i, lanes 0-15 -> M=i, N=lane; lanes 16-31 -> M=i+8.
    const int lane  = tid & 31;
    const int wv    = tid >> 5;            // 8 waves -> 8 (rb,cb) 16x16 output tiles
    const int rb    = (wv >> 2) << 4;      // 0 or 16
    const int cb    = (wv & 3) << 4;       // 0,16,32,48
    const int n     = lane & 15;           // N for B / M for A
    const int k0    = (lane >> 4) << 1;    // this half-wave's first K in the 4-chunk

    v8f accX  = {};
    v8f accXX = {};
    #pragma unroll
    for (int kb = 0; kb < 8; ++kb) {       // relative input rows 0..31 (band zero past 29)
        const int r0 = kb * 4 + k0;        // rel rows r0, r0+1 for this lane
        v2f a, bx, bxx;
        a.x = (n <= r0     && n >= r0 - 14) ? 1.0f : 0.0f;
        a.y = (n <= r0 + 1 && n >= r0 - 13) ? 1.0f : 0.0f;
        int ra0 = rb + r0;     if (ra0 > IN_ROWS - 1) ra0 = IN_ROWS - 1;  // band==0 there
        int ra1 = rb + r0 + 1; if (ra1 > IN_ROWS - 1) ra1 = IN_ROWS - 1;
        bx.x  = smem[HX_OFF  + ra0 * H_STRIDE + cb + n];
        bx.y  = smem[HX_OFF  + ra1 * H_STRIDE + cb + n];
        bxx.x = smem[HXX_OFF + ra0 * H_STRIDE + cb + n];
        bxx.y = smem[HXX_OFF + ra1 * H_STRIDE + cb + n];
        accX  = __builtin_amdgcn_wmma_f32_16x16x4_f32(false, a, false, bx,
                                                      (short)0, accX,  false, false);
        accXX = __builtin_amdgcn_wmma_f32_16x16x4_f32(false, a, false, bxx,
                                                      (short)0, accXX, false, false);
    }

    // ---- Normalize + coalesced store (lanes 0-15 one row, 16-31 the row+8).
    const float inv225 = 1.0f / 225.0f;
    const int   half8  = (lane >> 4) << 3;
    #pragma unroll
    for (int i = 0; i < 8; ++i) {
        const int ro = rb + i + half8;     // output row within 32-tall tile
        const int co = cb + n;             // output col within 64-wide tile
        const float mean = accX[i]  * inv225;
        const float msq  = accXX[i] * inv225;
        float var = msq - mean * mean;
        var = fmaxf(var, 1e-6f);
        const float xc  = smem[(ro + HALO) * TILE_STRIDE + co + 8];
        const float res = (xc - mean) / (sqrtf(var) + 1e-6f);
        op[(size_t)(row0 + ro) * IMG_W + (col0 + co)] = res;
    }
}

extern "C" void kernel_launch(void* const* d_in, const int* in_sizes, int n_in,
                              void* d_out, int out_size, void* d_ws, size_t ws_size,
                              hipStream_t stream) {
    const float* x = (const float*)d_in[0];
    float* out     = (float*)d_out;
    dim3 grid(IMG_W / TILE_W, IMG_W / TILE_H, 24);   // 16 x 32 x (8*3) planes
    dim3 block(256);
    const size_t shmem = SMEM_DW * sizeof(float);    // 39376 B
    lce_box15_kernel<<<grid, block, shmem, stream>>>(x, out);
}